// osclsNG_perinstance_top20_1245540516269
// MI455X (gfx1250) — compile-verified
//
#include <hip/hip_runtime.h>
#include <stdint.h>

// ---------------------------------------------------------------------------
// osclsNG per-instance top-20 CE on MI455X (gfx1250)
//
// Memory-bound streaming top-k: 512MB read once -> ~22us at 23.3 TB/s.
// No GEMM => WMMA not applicable. CDNA5 paths used instead:
//   * GLOBAL_LOAD_ASYNC_TO_LDS_B128 double-buffered staging (ASYNCcnt)
//   * s_wait_asynccnt pipelined waits
//   * wave32 distributed top-20: list entry i lives in lane i (lanes 0..19),
//     tau = exact running 20th-largest via lane-19 broadcast; insertions are
//     ballot/leader/shfl_up shift-insert (~12 ops each, ~140 events/row).
// ---------------------------------------------------------------------------

namespace {
constexpr int   N_ROWS   = 16384;
constexpr int   V_COLS   = 8192;
constexpr int   NUM_INST = 1024;
constexpr int   TOPK     = 20;
constexpr float NEG_FILL = -9999.0f;
constexpr float NEG_INF  = -3.0e38f;

constexpr int WPB    = 8;               // waves (== rows) per block
constexpr int CHUNK  = 512;             // floats per staging buffer (2KB)
constexpr int NCHUNK = V_COLS / CHUNK;  // 16
constexpr int SUBS   = CHUNK / 128;     // 4 float4-groups of 128 floats/wave

constexpr int LDS_BYTES = WPB * 2 * CHUNK * 4;   // 32KB staging only
}  // namespace

#define AS1 __attribute__((address_space(1)))
#define AS3 __attribute__((address_space(3)))

typedef int v4i __attribute__((vector_size(16)));

// ---- CDNA5 async global->LDS copy (16B per lane per issue) -----------------
__device__ __forceinline__ void async_cp_b128(const float* g, float* l) {
#if __has_builtin(__builtin_amdgcn_global_load_async_to_lds_b128)
  __builtin_amdgcn_global_load_async_to_lds_b128(
      (AS1 v4i*)g, (AS3 v4i*)l, /*imm offset*/ 0, /*cpol*/ 0);
#else
  unsigned lds_addr = (unsigned)(unsigned long long)(AS3 void*)l;
  unsigned long long ga = (unsigned long long)g;
  asm volatile("global_load_async_to_lds_b128 %0, %1, off"
               :: "v"(lds_addr), "v"(ga) : "memory");
#endif
}

#if __has_builtin(__builtin_amdgcn_s_wait_asynccnt)
#define WAIT_ASYNC(n) do { __builtin_amdgcn_s_wait_asynccnt(n); \
                           asm volatile("" ::: "memory"); } while (0)
#else
#define WAIT_ASYNC(n) asm volatile("s_wait_asynccnt %0" :: "i"(n) : "memory")
#endif

__global__ __launch_bounds__(256) void topk_ce_kernel(
    const float* __restrict__ outcls,
    const int*   __restrict__ label,
    const int*   __restrict__ instmap,
    float*       __restrict__ sums,   // [NUM_INST]
    float*       __restrict__ cnts)   // [NUM_INST]
{
  extern __shared__ float smem[];
  const int lane = threadIdx.x & 31;
  const int wave = threadIdx.x >> 5;
  const int row  = blockIdx.x * WPB + wave;

  float* stage = smem + wave * (2 * CHUNK);

  const float* rowp = outcls + (size_t)row * V_COLS;
  const int   lab   = label[row];
  const float tlog  = (lab >= 0) ? rowp[lab] : 0.0f;

  // Where does the target column land in our (chunk, sub, lane, elem) layout?
  const int labChunk = (lab >= 0) ? (lab >> 9) : -1;   // /512
  const int labSub   = (lab >> 7) & (SUBS - 1);        // /128 % 4
  const int labLane  = (lab >> 2) & 31;                // /4   % 32
  const int labElem  = lab & 3;

  // Distributed sorted top-20: entry i in lane i (lanes 0..19); tau = exact
  // running 20th-largest of the row. Values <= tau can never be in top-20.
  float gl  = NEG_INF;
  float tau = NEG_INF;

  // One candidate (wave-uniform loop; ~140 executions per row total).
#define INSERT_COMP(vc)                                           \
  {                                                               \
    float vj = (vc);                                              \
    for (;;) {                                                    \
      unsigned long long b = __ballot(vj > tau);                  \
      if (!b) break;                                              \
      int leader = __ffsll(b) - 1;                                \
      float x = __shfl(vj, leader);                               \
      if (lane == leader) vj = NEG_INF;  /* consume */            \
      float prev = __shfl_up(gl, 1);                              \
      bool  keep = (gl >= x);                                     \
      bool  pk   = (lane == 0) || (prev >= x);                    \
      gl  = keep ? gl : (pk ? x : prev);                          \
      tau = __shfl(gl, TOPK - 1);                                 \
    }                                                             \
  }

  // Prologue: stage chunks 0 and 1 (4 async b128 issues each).
#pragma unroll
  for (int i = 0; i < 4; ++i)
    async_cp_b128(rowp + i * 128 + lane * 4, stage + i * 128 + lane * 4);
#pragma unroll
  for (int i = 0; i < 4; ++i)
    async_cp_b128(rowp + CHUNK + i * 128 + lane * 4,
                  stage + CHUNK + i * 128 + lane * 4);

  for (int c = 0; c < NCHUNK; ++c) {
    // Retire the oldest 4 asyncs (chunk c); keep chunk c+1 in flight.
    if (c + 1 < NCHUNK) { WAIT_ASYNC(4); } else { WAIT_ASYNC(0); }

    const float* buf = stage + (c & 1) * CHUNK;
#pragma unroll
    for (int s = 0; s < SUBS; ++s) {
      float4 v = *(const float4*)(buf + s * 128 + lane * 4);  // ds_load_b128
      if (c == labChunk && s == labSub && lane == labLane) {
        if      (labElem == 0) v.x = NEG_FILL;
        else if (labElem == 1) v.y = NEG_FILL;
        else if (labElem == 2) v.z = NEG_FILL;
        else                   v.w = NEG_FILL;
      }
      float m4 = fmaxf(fmaxf(v.x, v.y), fmaxf(v.z, v.w));
      // Fast path: nothing in this 128-value group beats the exact row 20th.
      if (__ballot(m4 > tau)) {
        INSERT_COMP(v.x); INSERT_COMP(v.y);
        INSERT_COMP(v.z); INSERT_COMP(v.w);
      }
    }

    // Issue chunk c+2 into the buffer we just finished consuming.
    if (c + 2 < NCHUNK) {
      const float* g = rowp + (c + 2) * CHUNK + lane * 4;
      float* l = stage + (c & 1) * CHUNK + lane * 4;
#pragma unroll
      for (int i = 0; i < 4; ++i)
        async_cp_b128(g + i * 128, l + i * 128);
    }
  }

  // Lanes 0..19 now hold the exact row top-20 (descending). Logsumexp of
  // [tlog, top20]; loss = M + log(sum exp(x - M)) - tlog.
  float top1 = __shfl(gl, 0);
  float M    = fmaxf(top1, tlog);
  float e    = (lane < TOPK) ? expf(gl - M) : 0.0f;
#pragma unroll
  for (int off = 16; off >= 1; off >>= 1) e += __shfl_xor(e, off);

  if (lane == 0) {
    float ssum = e + expf(tlog - M);
    float loss = (lab >= 0) ? (M + logf(ssum) - tlog) : 0.0f;
    const int inst = instmap[row];
    atomicAdd(&sums[inst], loss);
    atomicAdd(&cnts[inst], 1.0f);   // reference counts every row
  }
#undef INSERT_COMP
}

__global__ void zero_kernel(float* p, int n) {
  int i = blockIdx.x * blockDim.x + threadIdx.x;
  if (i < n) p[i] = 0.0f;
}

__global__ void finalize_kernel(const float* __restrict__ sums,
                                const float* __restrict__ cnts,
                                float* __restrict__ out) {
  int i = blockIdx.x * blockDim.x + threadIdx.x;
  if (i < NUM_INST) {
    float c = cnts[i];
    out[i] = (c > 0.0f) ? sums[i] / fmaxf(c, 1.0f) : 0.0f;
  }
}

extern "C" void kernel_launch(void* const* d_in, const int* in_sizes, int n_in,
                              void* d_out, int out_size, void* d_ws, size_t ws_size,
                              hipStream_t stream) {
  (void)in_sizes; (void)n_in; (void)out_size; (void)ws_size;
  const float* outcls  = (const float*)d_in[0];
  const int*   label   = (const int*)d_in[1];
  const int*   instmap = (const int*)d_in[2];
  float* out  = (float*)d_out;
  float* sums = (float*)d_ws;            // NUM_INST floats
  float* cnts = sums + NUM_INST;         // NUM_INST floats

  zero_kernel<<<(2 * NUM_INST + 255) / 256, 256, 0, stream>>>(sums, 2 * NUM_INST);
  topk_ce_kernel<<<N_ROWS / WPB, 256, LDS_BYTES, stream>>>(
      outcls, label, instmap, sums, cnts);
  finalize_kernel<<<(NUM_INST + 255) / 256, 256, 0, stream>>>(sums, cnts, out);
}